// GraphAttentionLayer_10977936409197
// MI455X (gfx1250) — compile-verified
//
#include <hip/hip_runtime.h>
#include <hip/hip_bf16.h>

// GAT layer, MI455X (gfx1250), wave32.
// d_in: h[8192*512] f32, W[512*64] f32, a[128] f32, adj[8192*8192] i32, adj_new[8192*8192] i32
// d_out: out[8192*64] f32 ++ attention[8192*8192] f32

typedef float v2f __attribute__((ext_vector_type(2)));
typedef float v8f __attribute__((ext_vector_type(8)));

#define NREF     8192
#define FIN      512
#define FOUT     64
#define MWORDS   (NREF / 32)      // 256 mask words per row
#define NEG_BIG  (-9.0e15f)
#define ALPHA    0.2f

__device__ __forceinline__ float lrelu(float x) { return x > 0.0f ? x : ALPHA * x; }
__device__ __forceinline__ float eluf(float x)  { return x > 0.0f ? x : __expf(x) - 1.0f; }

// ---------------------------------------------------------------------------
// Kernel 1: Wh = h @ W   (8192x512 @ 512x64), fp32 WMMA 16x16x4.
// One wave per 16x16 output tile. 512 row-tiles x 4 col-tiles = 2048 waves.
// A-frag (16x4 f32): lane L -> row m=L&15, VGPR v holds K = 2*(L>>4)+v.
// B-frag (4x16 f32): lane L -> col n=L&15, VGPR v holds K = 2*(L>>4)+v.
// C/D (16x16 f32):   lane L -> col n=L&15, VGPR r holds row r + 8*(L>>4).
// ---------------------------------------------------------------------------
__global__ __launch_bounds__(256) void k_wh(const float* __restrict__ h,
                                            const float* __restrict__ W,
                                            float* __restrict__ Wh)
{
    const int gwave = (blockIdx.x * blockDim.x + threadIdx.x) >> 5;  // 0..2047
    const int lane  = threadIdx.x & 31;
    const int rt = gwave >> 2;          // row tile 0..511
    const int ct = gwave & 3;           // col tile 0..3
    const int i0 = rt * 16;
    const int n0 = ct * 16;
    const int m    = lane & 15;
    const int hi   = lane >> 4;
    const int koff = hi * 2;

    const float* __restrict__ hrow = h + (size_t)(i0 + m) * FIN;

    v8f acc = {};
    for (int k = 0; k < FIN; k += 4) {
        v2f a;
        {   // two consecutive k's -> one 64-bit load
            float2 t = *(const float2*)(hrow + k + koff);
            a.x = t.x; a.y = t.y;
        }
        v2f b;
        b.x = W[(size_t)(k + koff)     * FOUT + n0 + m];
        b.y = W[(size_t)(k + koff + 1) * FOUT + n0 + m];
        acc = __builtin_amdgcn_wmma_f32_16x16x4_f32(false, a, false, b,
                                                    (short)0, acc, false, false);
    }
#pragma unroll
    for (int r = 0; r < 8; ++r) {
        const int row = i0 + r + 8 * hi;
        Wh[(size_t)row * FOUT + n0 + m] = acc[r];
    }
}

// ---------------------------------------------------------------------------
// Kernel 2: Wh1[i] = Wh[i,:] . a[0:64],  Wh2[i] = Wh[i,:] . a[64:128]
// ---------------------------------------------------------------------------
__global__ __launch_bounds__(256) void k_wh12(const float* __restrict__ Wh,
                                              const float* __restrict__ a,
                                              float* __restrict__ Wh1,
                                              float* __restrict__ Wh2)
{
    const int i = blockIdx.x * blockDim.x + threadIdx.x;
    if (i >= NREF) return;
    const float* __restrict__ row = Wh + (size_t)i * FOUT;
    float s1 = 0.0f, s2 = 0.0f;
#pragma unroll
    for (int c = 0; c < FOUT; ++c) {
        const float v = row[c];
        s1 += v * a[c];
        s2 += v * a[FOUT + c];
    }
    Wh1[i] = s1;
    Wh2[i] = s2;
}

// ---------------------------------------------------------------------------
// Kernel 3: per-row masked softmax stats for both masks + adjacency bitmask
// compression (wave32 ballot -> 1 bit per edge, 32x traffic cut for kernel 4).
// One 256-thread workgroup (8 waves) per row.
//   pass 1: row max (both masks) + ballot-compress adj rows to mask words
//   pass 2: sum(exp(x - max))  (adj rows still L2-resident)
// stats[i*4] = {max_m, sum_m, max_n, sum_n}
// ---------------------------------------------------------------------------
__global__ __launch_bounds__(256) void k_stats(const int* __restrict__ adj,
                                               const int* __restrict__ adjn,
                                               const float* __restrict__ Wh1,
                                               const float* __restrict__ Wh2,
                                               float* __restrict__ stats,
                                               unsigned int* __restrict__ maskm,
                                               unsigned int* __restrict__ maskn)
{
    __shared__ float red[512];
    const int i    = blockIdx.x;
    const int tid  = threadIdx.x;
    const int wv   = tid >> 5;
    const int lane = tid & 31;
    const float w1 = Wh1[i];
    const int* __restrict__ ar  = adj  + (size_t)i * NREF;
    const int* __restrict__ anr = adjn + (size_t)i * NREF;
    unsigned int* __restrict__ mr = maskm + (size_t)i * MWORDS;
    unsigned int* __restrict__ nr = maskn + (size_t)i * MWORDS;

    // pass 1: row max for both masks + bitmask compression
    float mm = -3.0e38f, mn = -3.0e38f;
    for (int jb = wv * 32; jb < NREF; jb += 256) {
        const int j = jb + lane;                      // coalesced 128B per wave
        __builtin_prefetch(ar + j + 2048, 0, 0);
        __builtin_prefetch(anr + j + 2048, 0, 0);
        const float e = lrelu(w1 + Wh2[j]);
        const bool bm = ar[j]  > 0;
        const bool bn = anr[j] > 0;
        const unsigned int vm = (unsigned int)__ballot(bm);   // wave32: low 32 bits
        const unsigned int vn = (unsigned int)__ballot(bn);
        if (lane == 0) {
            mr[jb >> 5] = vm;
            nr[jb >> 5] = vn;
        }
        mm = fmaxf(mm, bm ? e : NEG_BIG);
        mn = fmaxf(mn, bn ? e : NEG_BIG);
    }
    red[tid] = mm; red[256 + tid] = mn;
    __syncthreads();
    for (int off = 128; off > 0; off >>= 1) {
        if (tid < off) {
            red[tid]       = fmaxf(red[tid],       red[tid + off]);
            red[256 + tid] = fmaxf(red[256 + tid], red[256 + tid + off]);
        }
        __syncthreads();
    }
    mm = red[0]; mn = red[256];
    __syncthreads();

    // pass 2: sum of exp (adj rows now in L2)
    float sm = 0.0f, sn = 0.0f;
    for (int j = tid; j < NREF; j += 256) {
        const float e  = lrelu(w1 + Wh2[j]);
        const float xm = (ar[j]  > 0) ? e : NEG_BIG;
        const float xn = (anr[j] > 0) ? e : NEG_BIG;
        sm += __expf(xm - mm);
        sn += __expf(xn - mn);
    }
    red[tid] = sm; red[256 + tid] = sn;
    __syncthreads();
    for (int off = 128; off > 0; off >>= 1) {
        if (tid < off) {
            red[tid]       += red[tid + off];
            red[256 + tid] += red[256 + tid + off];
        }
        __syncthreads();
    }
    if (tid == 0) {
        stats[(size_t)i * 4 + 0] = mm;
        stats[(size_t)i * 4 + 1] = red[0];
        stats[(size_t)i * 4 + 2] = mn;
        stats[(size_t)i * 4 + 3] = red[256];
    }
}

// ---------------------------------------------------------------------------
// Kernel 4: fused attention materialization + h_prime = attention @ Wh + ELU.
// One block per 16 rows; 8 waves split K=8192 (stride 32 columns each).
// Adjacency comes from the compressed bitmasks: per iteration each lane reads
// ONE mask word per matrix (word index == iteration count; bit position
// wv*4+koff is loop-invariant). Each lane computes its 2 WMMA-A-fragment
// attention values directly (row m=lane&15, cols kb+2*(lane>>4)+{0,1}),
// streams them to the attention output, and feeds 4 fp32 WMMAs (one per
// 16-col tile of the 64-wide output). Split-K partials reduced through LDS.
// ---------------------------------------------------------------------------
__global__ __launch_bounds__(256) void k_att_gemm(const unsigned int* __restrict__ maskm,
                                                  const unsigned int* __restrict__ maskn,
                                                  const float* __restrict__ Wh,
                                                  const float* __restrict__ Wh1,
                                                  const float* __restrict__ Wh2,
                                                  const float* __restrict__ stats,
                                                  float* __restrict__ out,
                                                  float* __restrict__ att)
{
    __shared__ float red[8192];   // 8 waves * (4 tiles * 32 lanes * 8 regs)

    const int i0   = blockIdx.x * 16;
    const int wv   = threadIdx.x >> 5;
    const int lane = threadIdx.x & 31;
    const int m    = lane & 15;
    const int hi   = lane >> 4;
    const int koff = hi * 2;
    const int row  = i0 + m;
    const int bit0 = wv * 4 + koff;      // loop-invariant bit position (<= 30)

    const float w1   = Wh1[row];
    const float maxm = stats[(size_t)row * 4 + 0];
    const float rsm  = 1.0f / stats[(size_t)row * 4 + 1];
    const float maxn = stats[(size_t)row * 4 + 2];
    const float rsn  = 1.0f / stats[(size_t)row * 4 + 3];

    const unsigned int* __restrict__ mr = maskm + (size_t)row * MWORDS;
    const unsigned int* __restrict__ nr = maskn + (size_t)row * MWORDS;
    float* __restrict__ atr = att + (size_t)row * NREF;

    v8f acc[4] = {v8f{}, v8f{}, v8f{}, v8f{}};

    int it = 0;
    for (int kb = wv * 4; kb < NREF; kb += 32, ++it) {
        const int j = kb + koff;
        const unsigned int wm = mr[it];
        const unsigned int wn = nr[it];
        const float2 w2 = *(const float2*)(Wh2 + j);

        const float e0  = lrelu(w1 + w2.x);
        const float e1  = lrelu(w1 + w2.y);
        const float xm0 = ((wm >> bit0)       & 1u) ? e0 : NEG_BIG;
        const float xm1 = ((wm >> (bit0 + 1)) & 1u) ? e1 : NEG_BIG;
        const float xn0 = ((wn >> bit0)       & 1u) ? e0 : NEG_BIG;
        const float xn1 = ((wn >> (bit0 + 1)) & 1u) ? e1 : NEG_BIG;

        const float a0 = __expf(xm0 - maxm) * rsm + __expf(xn0 - maxn) * rsn;
        const float a1 = __expf(xm1 - maxm) * rsm + __expf(xn1 - maxn) * rsn;

        *(float2*)(atr + j) = make_float2(a0, a1);   // attention output

        v2f afrag; afrag.x = a0; afrag.y = a1;

        const float* __restrict__ wr0 = Wh + (size_t)j * FOUT;  // rows j, j+1 of Wh
        const float* __restrict__ wr1 = wr0 + FOUT;
#pragma unroll
        for (int t = 0; t < 4; ++t) {
            v2f b;
            b.x = wr0[t * 16 + m];
            b.y = wr1[t * 16 + m];
            acc[t] = __builtin_amdgcn_wmma_f32_16x16x4_f32(false, afrag, false, b,
                                                           (short)0, acc[t], false, false);
        }
    }

    // split-K reduction across the 8 waves
    float* my = red + wv * 1024;
#pragma unroll
    for (int t = 0; t < 4; ++t)
#pragma unroll
        for (int r = 0; r < 8; ++r)
            my[t * 256 + lane * 8 + r] = acc[t][r];
    __syncthreads();

#pragma unroll
    for (int q = 0; q < 4; ++q) {
        const int f = threadIdx.x * 4 + q;      // flat index over 4*32*8 = 1024 values
        float s = 0.0f;
#pragma unroll
        for (int w = 0; w < 8; ++w) s += red[w * 1024 + f];
        const int t = f >> 8;
        const int l = (f >> 3) & 31;
        const int r = f & 7;
        const int orow = i0 + r + 8 * (l >> 4);
        const int ocol = t * 16 + (l & 15);
        out[(size_t)orow * FOUT + ocol] = eluf(s);
    }
}

// ---------------------------------------------------------------------------
extern "C" void kernel_launch(void* const* d_in, const int* in_sizes, int n_in,
                              void* d_out, int out_size, void* d_ws, size_t ws_size,
                              hipStream_t stream)
{
    const float* h    = (const float*)d_in[0];
    const float* W    = (const float*)d_in[1];
    const float* a    = (const float*)d_in[2];
    const int*   adj  = (const int*)d_in[3];
    const int*   adjn = (const int*)d_in[4];

    float* out = (float*)d_out;                       // [8192*64]
    float* att = out + (size_t)NREF * FOUT;           // [8192*8192]

    float* ws    = (float*)d_ws;
    float* Wh    = ws;                                // 8192*64
    float* Wh1   = Wh  + (size_t)NREF * FOUT;         // 8192
    float* Wh2   = Wh1 + NREF;                        // 8192
    float* stats = Wh2 + NREF;                        // 8192*4
    unsigned int* maskm = (unsigned int*)(stats + (size_t)NREF * 4);  // 8192*256 words
    unsigned int* maskn = maskm + (size_t)NREF * MWORDS;              // 8192*256 words

    // 1) Wh = h @ W            (2048 waves -> 256 blocks of 8 waves)
    k_wh<<<256, 256, 0, stream>>>(h, W, Wh);
    // 2) Wh1 / Wh2
    k_wh12<<<NREF / 256, 256, 0, stream>>>(Wh, a, Wh1, Wh2);
    // 3) per-row softmax stats + adjacency bitmask compression
    k_stats<<<NREF, 256, 0, stream>>>(adj, adjn, Wh1, Wh2, stats, maskm, maskn);
    // 4) attention + h_prime GEMM + ELU (reads bitmasks, not adjacency)
    k_att_gemm<<<NREF / 16, 256, 0, stream>>>(maskm, maskn, Wh, Wh1, Wh2, stats, out, att);
}